// PEABaseChannel_5652176961550
// MI455X (gfx1250) — compile-verified
//
#include <hip/hip_runtime.h>
#include <hip/hip_bf16.h>

// 2-layer GCN (PyG GCNConv semantics) for MI455X / gfx1250.
// Factorization: g[i] = (x@W)[i] * dinv[i];  out[d] = dinv[d]*(g[d] + sum_{e:dst=d} g[src_e]) + b
// GEMM uses V_WMMA_F32_16X16X4_F32 (wave32, 16x16 tile per wave, K chained in 4s).

typedef __attribute__((ext_vector_type(2))) float v2f;
typedef __attribute__((ext_vector_type(8))) float v8f;

#define FD 64  // feature dim

// deg[i] = 1 (self loop contribution)
__global__ void k_init_deg(float* __restrict__ deg, int N) {
  int i = blockIdx.x * blockDim.x + threadIdx.x;
  if (i < N) deg[i] = 1.0f;
}

// deg[dst_e] += 1 per edge
__global__ void k_edge_deg(const int* __restrict__ dst, float* __restrict__ deg, int E) {
  int i = blockIdx.x * blockDim.x + threadIdx.x;
  if (i < E) atomicAdd(&deg[dst[i]], 1.0f);
}

// deg -> rsqrt(deg) in place (deg >= 1 always, no max needed)
__global__ void k_dinv(float* __restrict__ deg, int N) {
  int i = blockIdx.x * blockDim.x + threadIdx.x;
  if (i < N) deg[i] = rsqrtf(deg[i]);
}

// g[i,:] = (X[i,:] @ W) * dinv[i]; acc seeded with g (self-loop term).
// One wave per 16x16 output tile; W staged in LDS (16 KB) per block.
// Fragment layouts per CDNA5 ISA 7.12.2:
//   A (16x4 f32):  lanes 0-15 rows M, VGPR{0,1} = K{0,1}; lanes 16-31 = K{2,3}
//   B (4x16 f32):  VGPR v holds row K=v (lanes 0-15) and K=v+2 (lanes 16-31), cols N=lane&15
//   C/D (16x16):   VGPR v holds row m0+v (lanes 0-15) and m0+8+v (lanes 16-31)
__global__ void __launch_bounds__(128)
k_gemm_scale_wmma(const float* __restrict__ X, const float* __restrict__ W,
                  const float* __restrict__ dinv,
                  float* __restrict__ g, float* __restrict__ acc, int N) {
  __shared__ float Ws[FD * FD];
  for (int i = threadIdx.x; i < FD * FD; i += blockDim.x) Ws[i] = W[i];
  __syncthreads();

  const int lane = threadIdx.x & 31;
  const int wave = threadIdx.x >> 5;
  const int tile = blockIdx.x * 4 + wave;           // global wave id
  const int rowTiles = (N + 15) >> 4;
  if (tile >= rowTiles * 4) return;                 // wave-uniform exit (EXEC stays full for WMMA)

  const int m0 = (tile >> 2) << 4;                  // row tile base
  const int n0 = (tile & 3) << 4;                   // col tile base (FD/16 == 4)
  const int half = lane >> 4;                       // 0: lanes 0-15, 1: lanes 16-31
  const int l15 = lane & 15;

  int rowA = m0 + l15;
  if (rowA >= N) rowA = N - 1;                      // clamp keeps EXEC full; junk rows never stored
  const float* xr = X + (size_t)rowA * FD;

  v8f c = {};
#pragma unroll
  for (int k = 0; k < FD; k += 4) {
    v2f a, b;
    a.x = xr[k + 2 * half + 0];
    a.y = xr[k + 2 * half + 1];
    b.x = Ws[(k + 2 * half + 0) * FD + n0 + l15];
    b.y = Ws[(k + 2 * half + 1) * FD + n0 + l15];
    // (neg_a, A, neg_b, B, c_mod, C, reuse_a, reuse_b)
    c = __builtin_amdgcn_wmma_f32_16x16x4_f32(false, a, false, b, (short)0, c, false, false);
  }

#pragma unroll
  for (int v = 0; v < 8; ++v) {
    int row = m0 + v + 8 * half;
    if (row < N) {
      float val = c[v] * dinv[row];
      size_t o = (size_t)row * FD + n0 + l15;
      g[o] = val;    // normalized features, gathered by edge kernel
      acc[o] = val;  // accumulator seeded with self-loop term
    }
  }
}

// Per edge-feature scatter: acc[dst,f] += g[src,f]. 64 threads per edge.
// g (25.6 MB) lives in the 192 MB L2; atomics resolve at L2.
__global__ void k_edge_acc(const int* __restrict__ src, const int* __restrict__ dst,
                           const float* __restrict__ g, float* __restrict__ acc, int E) {
  long long t = (long long)blockIdx.x * blockDim.x + threadIdx.x;
  int e = (int)(t >> 6);
  int f = (int)(t & 63);
  if (e < E) {
    int s = src[e];
    int d = dst[e];
    atomicAdd(&acc[(size_t)d * FD + f], g[(size_t)s * FD + f]);
  }
}

// out[i,f] = acc[i,f] * dinv[i] + b[f], optional ReLU
__global__ void k_finalize(const float* __restrict__ acc, const float* __restrict__ dinv,
                           const float* __restrict__ bias, float* __restrict__ out,
                           long long total, int relu) {
  long long t = (long long)blockIdx.x * blockDim.x + threadIdx.x;
  if (t < total) {
    int i = (int)(t >> 6);
    int f = (int)(t & 63);
    float v = acc[t] * dinv[i] + bias[f];
    if (relu) v = fmaxf(v, 0.0f);
    out[t] = v;
  }
}

extern "C" void kernel_launch(void* const* d_in, const int* in_sizes, int n_in,
                              void* d_out, int out_size, void* d_ws, size_t ws_size,
                              hipStream_t stream) {
  const float* x  = (const float*)d_in[0];
  const float* W1 = (const float*)d_in[1];
  const float* b1 = (const float*)d_in[2];
  const float* W2 = (const float*)d_in[3];
  const float* b2 = (const float*)d_in[4];
  const int*   e0 = (const int*)d_in[5];   // [2, E0] flat: src at 0, dst at E0
  const int*   e1 = (const int*)d_in[6];

  const int N  = in_sizes[0] / FD;
  const int E0 = in_sizes[5] / 2;
  const int E1 = in_sizes[6] / 2;
  float* out = (float*)d_out;

  // Workspace: dinv[N] | g[N*FD] | acc[N*FD]  (~51.6 MB for N=100k)
  float* dinv = (float*)d_ws;
  float* g    = dinv + N;
  float* acc  = g + (size_t)N * FD;

  const int TPB = 256;

  auto layer = [&](const float* F, const float* W, const float* bias,
                   const int* ei, int E, int relu, float* O) {
    const int* esrc = ei;
    const int* edst = ei + E;

    k_init_deg<<<(N + TPB - 1) / TPB, TPB, 0, stream>>>(dinv, N);
    k_edge_deg<<<(E + TPB - 1) / TPB, TPB, 0, stream>>>(edst, dinv, E);
    k_dinv<<<(N + TPB - 1) / TPB, TPB, 0, stream>>>(dinv, N);

    int rowTiles = (N + 15) / 16;
    int waves = rowTiles * 4;                 // 4 col tiles (FD=64)
    k_gemm_scale_wmma<<<(waves + 3) / 4, 128, 0, stream>>>(F, W, dinv, g, acc, N);

    long long workAcc = (long long)E * FD;
    k_edge_acc<<<(unsigned)((workAcc + TPB - 1) / TPB), TPB, 0, stream>>>(esrc, edst, g, acc, E);

    long long workFin = (long long)N * FD;
    k_finalize<<<(unsigned)((workFin + TPB - 1) / TPB), TPB, 0, stream>>>(acc, dinv, bias, O, workFin, relu);
  };

  // Layer 1: conv + ReLU -> d_out; layer 2 GEMM consumes d_out into g before
  // finalize overwrites d_out (stream-ordered), so no extra buffer needed.
  layer(x,   W1, b1, e0, E0, /*relu=*/1, out);
  layer(out, W2, b2, e1, E1, /*relu=*/0, out);
}